// SentenceProbability_88699664597346
// MI455X (gfx1250) — compile-verified
//
#include <hip/hip_runtime.h>
#include <math.h>

typedef __attribute__((ext_vector_type(2))) float v2f;
typedef __attribute__((ext_vector_type(8))) float v8f;

#define D_DIM 100
#define KC 25      // D/4 k-chunks for V_WMMA_F32_16X16X4_F32
#define NT 7       // ceil(D/16) n-tiles
#define GN 112     // NT*16 padded row stride of G

// ---------------------------------------------------------------------------
// K1: ds[b,e] = sum_d d[b,d] * w_s[d,e]   (tiny: 128x100x100)
// ---------------------------------------------------------------------------
__global__ void k_ds(const float* __restrict__ dv, const float* __restrict__ w_s,
                     float* __restrict__ ds) {
  int b = blockIdx.x, e = threadIdx.x;
  if (e < D_DIM) {
    const float* db = dv + b * D_DIM;
    float acc = 0.f;
    #pragma unroll 4
    for (int k = 0; k < D_DIM; ++k) acc = fmaf(db[k], w_s[k * D_DIM + e], acc);
    ds[b * D_DIM + e] = acc;
  }
}

// ---------------------------------------------------------------------------
// K2: ap[t] = pos_table[t,:] . w_ap   (tiny)
// ---------------------------------------------------------------------------
__global__ void k_abs(const float* __restrict__ pos, const float* __restrict__ w_ap,
                      float* __restrict__ ap, int T) {
  int t = blockIdx.x * blockDim.x + threadIdx.x;
  if (t < T) {
    const float* pt = pos + (size_t)t * D_DIM;
    float acc = 0.f;
    #pragma unroll 4
    for (int d = 0; d < D_DIM; ++d) acc = fmaf(pt[d], w_ap[d], acc);
    ap[t] = acc;
  }
}

// ---------------------------------------------------------------------------
// K3: G[m,e] = sum_d h[m,d] * w_r[e,d]  (M = B*T, fp32 WMMA 16x16x4)
// One wave computes a 16(M) x 112(N) strip; 8 waves / block.
// w_r^T fragments pre-swizzled into LDS in the wmma B-operand lane layout:
//   VGPR0: lanes 0-15 = B[k][n0+l], lanes 16-31 = B[k+2][n0+l-16]
//   VGPR1: same with k+1 / k+3  -> per-lane contiguous float2 in LDS.
// ---------------------------------------------------------------------------
__global__ void __launch_bounds__(256) k_gemm(const float* __restrict__ h,
                                              const float* __restrict__ w_r,
                                              float* __restrict__ G, int M) {
  __shared__ v2f Bf[KC][NT][32];
  const int tid  = threadIdx.x;
  const int lane = tid & 31;
  const int wave = tid >> 5;

  for (int i = tid; i < KC * NT * 32; i += 256) {
    int l  = i & 31;
    int nt = (i >> 5) % NT;
    int kc = (i >> 5) / NT;
    int n  = nt * 16 + (l & 15);
    int k  = kc * 4 + ((l & 16) ? 2 : 0);      // k, k+1 both < 100 always
    v2f v;
    v.x = (n < D_DIM) ? w_r[n * D_DIM + k]     : 0.f;
    v.y = (n < D_DIM) ? w_r[n * D_DIM + k + 1] : 0.f;
    Bf[kc][nt][l] = v;
  }
  __syncthreads();

  const int strip = blockIdx.x * 8 + wave;
  int row  = strip * 16 + (lane & 15);
  int rowc = row < M ? row : (M - 1);          // clamp (branchless, EXEC stays full)
  const float* arow = h + (size_t)rowc * D_DIM + ((lane & 16) ? 2 : 0);

  v8f c[NT];
  #pragma unroll
  for (int nt = 0; nt < NT; ++nt) c[nt] = (v8f){0.f,0.f,0.f,0.f,0.f,0.f,0.f,0.f};

  #pragma unroll
  for (int kc = 0; kc < KC; ++kc) {
    v2f a = *(const v2f*)(arow + kc * 4);      // A frag: {k,k+1} (lanes>=16: {k+2,k+3})
    #pragma unroll
    for (int nt = 0; nt < NT; ++nt) {
      c[nt] = __builtin_amdgcn_wmma_f32_16x16x4_f32(
          /*neg_a=*/false, a, /*neg_b=*/false, Bf[kc][nt][lane],
          /*c_mod=*/(short)0, c[nt], /*reuse_a=*/false, /*reuse_b=*/false);
    }
  }

  // C/D layout: VGPR r -> lanes 0-15: M=r, lanes 16-31: M=r+8
  const int rbase = strip * 16 + ((lane & 16) ? 8 : 0);
  const int col0  = lane & 15;
  #pragma unroll
  for (int nt = 0; nt < NT; ++nt) {
    #pragma unroll
    for (int r = 0; r < 8; ++r) {
      int rg = rbase + r;
      if (rg < M) G[(size_t)rg * GN + nt * 16 + col0] = c[nt][r];
    }
  }
}

// ---------------------------------------------------------------------------
// K4: sequential scan. One wave32 per batch element; lane holds 4 of the 100
// state elements (lanes 0..24 active; 100 = 25 * float4, 16B aligned).
// Per step: pre = h.(w_c+ds[b]) (fused), nov = tanh(s).G_j, p = sigmoid,
// s += h_j * p (masked at j==0). Next-step h/G loads software-pipelined.
// ---------------------------------------------------------------------------
__global__ void k_scan(const float* __restrict__ h, const float* __restrict__ G,
                       const float* __restrict__ w_c, const float* __restrict__ ds,
                       const float* __restrict__ ap, const float* __restrict__ bias,
                       float* __restrict__ out, int T) {
  const int b    = blockIdx.x;
  const int lane = threadIdx.x;
  const bool act = (lane * 4) < D_DIM;

  float4 cw = make_float4(0.f, 0.f, 0.f, 0.f);
  if (act) {
    float4 wc = *(const float4*)(w_c + lane * 4);
    float4 db = *(const float4*)(ds + b * D_DIM + lane * 4);
    cw = make_float4(wc.x + db.x, wc.y + db.y, wc.z + db.z, wc.w + db.w);
  }
  const float bval = bias[0];
  const float* hb = h + (size_t)b * T * D_DIM + lane * 4;
  const float* gb = G + (size_t)b * T * GN   + lane * 4;

  float4 s  = make_float4(0.f, 0.f, 0.f, 0.f);
  float4 h4 = make_float4(0.f, 0.f, 0.f, 0.f);
  float4 g4 = make_float4(0.f, 0.f, 0.f, 0.f);
  if (act) { h4 = *(const float4*)hb; g4 = *(const float4*)gb; }

  for (int j = 0; j < T; ++j) {
    float4 hn = make_float4(0.f, 0.f, 0.f, 0.f);
    float4 gn = make_float4(0.f, 0.f, 0.f, 0.f);
    if (act && (j + 1) < T) {
      hn = *(const float4*)(hb + (size_t)(j + 1) * D_DIM);
      gn = *(const float4*)(gb + (size_t)(j + 1) * GN);
      if (j + 6 < T) {                          // gfx1250 global_prefetch hints
        __builtin_prefetch(hb + (size_t)(j + 6) * D_DIM, 0, 1);
        __builtin_prefetch(gb + (size_t)(j + 6) * GN, 0, 1);
      }
    }

    float pre = h4.x * cw.x + h4.y * cw.y + h4.z * cw.z + h4.w * cw.w;
    float nov = tanhf(s.x) * g4.x + tanhf(s.y) * g4.y +
                tanhf(s.z) * g4.z + tanhf(s.w) * g4.w;
    #pragma unroll
    for (int off = 16; off > 0; off >>= 1) {    // wave32 reduction
      pre += __shfl_xor(pre, off, 32);
      nov += __shfl_xor(nov, off, 32);
    }

    float x = pre + ap[j] + bval - nov;
    float p = 1.f / (1.f + __expf(-x));
    if (j > 0) {                                // mask: no state update at j==0
      s.x = fmaf(h4.x, p, s.x);
      s.y = fmaf(h4.y, p, s.y);
      s.z = fmaf(h4.z, p, s.z);
      s.w = fmaf(h4.w, p, s.w);
    }
    if (lane == 0) out[(size_t)b * T + j] = p;
    h4 = hn; g4 = gn;
  }
}

// ---------------------------------------------------------------------------
extern "C" void kernel_launch(void* const* d_in, const int* in_sizes, int n_in,
                              void* d_out, int out_size, void* d_ws, size_t ws_size,
                              hipStream_t stream) {
  const float* h    = (const float*)d_in[0];
  const float* dv   = (const float*)d_in[1];
  const float* w_c  = (const float*)d_in[2];
  const float* w_s  = (const float*)d_in[3];
  const float* w_r  = (const float*)d_in[4];
  const float* pos  = (const float*)d_in[5];
  const float* w_ap = (const float*)d_in[6];
  const float* bias = (const float*)d_in[7];
  float* out = (float*)d_out;

  const int Bn = in_sizes[1] / D_DIM;            // d is [B,1,D]
  const int T  = in_sizes[0] / (Bn * D_DIM);     // h is [B,T,D]
  const int M  = Bn * T;

  float* G   = (float*)d_ws;                     // [M, 112]
  float* dsw = G + (size_t)M * GN;               // [B, 100]
  float* ap  = dsw + (size_t)Bn * D_DIM;         // [T]

  k_ds <<<Bn, 128, 0, stream>>>(dv, w_s, dsw);
  k_abs<<<(T + 255) / 256, 256, 0, stream>>>(pos, w_ap, ap, T);

  const int strips = (M + 15) / 16;
  k_gemm<<<(strips + 7) / 8, 256, 0, stream>>>(h, w_r, G, M);

  k_scan<<<Bn, 32, 0, stream>>>(h, G, w_c, dsw, ap, bias, out, T);
}